// MambaIntegerBlockV2_901943132630
// MI455X (gfx1250) — compile-verified
//
#include <hip/hip_runtime.h>
#include <hip/hip_bf16.h>

typedef __attribute__((ext_vector_type(2))) float v2f;
typedef __attribute__((ext_vector_type(8))) float v8f;

#define CS    64          // chunk size
#define HN    16          // heads
#define PD    64          // head dim P
#define SD    64          // state dim S
#define LSEQ  4096        // sequence length
#define NCH   (LSEQ/CS)   // 64 chunks
#define BATCH 2
#define LDL   68          // LDS row stride (floats), 16B-aligned rows

__device__ __forceinline__ v8f wmma4(v2f a, v2f b, v8f c) {
  // D = A(16x4 f32) * B(4x16 f32) + C(16x16 f32)
  return __builtin_amdgcn_wmma_f32_16x16x4_f32(
      /*neg_a=*/false, a, /*neg_b=*/false, b,
      /*c_mod=*/(short)0, c, /*reuse_a=*/false, /*reuse_b=*/false);
}

// ---------------------------------------------------------------------------
// Kernel 1: per (b,h,chunk) -> Y_intra (to d_out), h_final (to ws), Acs (to ws)
// ---------------------------------------------------------------------------
__global__ __launch_bounds__(128)
void ssd_chunk_kernel(const float* __restrict__ X, const float* __restrict__ A,
                      const float* __restrict__ Bm, const float* __restrict__ Cm,
                      float* __restrict__ Y, float* __restrict__ hf,
                      float* __restrict__ acsbuf)
{
  __shared__ float sC[CS * LDL];   // C tile; later reused for M = L o CB
  __shared__ float sB[CS * LDL];
  __shared__ float sX[CS * LDL];
  __shared__ float sAcs[CS];
  __shared__ float sDte[CS];

  const int blk    = blockIdx.x;       // 0..2047
  const int nchunk = blk & (NCH - 1);
  const int bh     = blk >> 6;         // 0..31
  const int h      = bh & (HN - 1);
  const int b      = bh >> 4;

  const int tid = threadIdx.x;
  const long row0   = (long)(b * LSEQ + nchunk * CS) * HN + h;
  const float* gX = X  + row0 * PD;
  const float* gB = Bm + row0 * SD;
  const float* gC = Cm + row0 * SD;
  const float* gA = A  + (long)(b * LSEQ + nchunk * CS) * HN + h;

  // stage 64x64 tiles (global row stride = HN*64 floats)
  for (int f = tid; f < CS * 16; f += 128) {
    const int row = f >> 4;
    const int c4  = (f & 15) << 2;
    const float4 xv = *(const float4*)(gX + (long)row * (HN * PD) + c4);
    const float4 bv = *(const float4*)(gB + (long)row * (HN * SD) + c4);
    const float4 cv = *(const float4*)(gC + (long)row * (HN * SD) + c4);
    *(float4*)(&sX[row * LDL + c4]) = xv;
    *(float4*)(&sB[row * LDL + c4]) = bv;
    *(float4*)(&sC[row * LDL + c4]) = cv;
  }
  if (tid < CS) sAcs[tid] = gA[(long)tid * HN];
  __syncthreads();
  if (tid == 0) {                      // serial cumsum over 64 values
    float s = 0.f;
    for (int i = 0; i < CS; ++i) { s += sAcs[i]; sAcs[i] = s; }
  }
  __syncthreads();
  if (tid < CS) {
    sDte[tid] = expf(sAcs[CS - 1] - sAcs[tid]);     // decay_to_end
    acsbuf[((long)bh * NCH + nchunk) * CS + tid] = sAcs[tid];
  }
  __syncthreads();

  const int lane  = tid & 31;
  const int wave  = tid >> 5;
  const int lrow  = lane & 15;            // m (A) / n (B,C,D) within tile
  const int khalf = (lane >> 4) << 1;     // k offset: 0 or 2
  const int mhalf = (lane >> 4) << 3;     // m offset in C/D: 0 or 8
  const int tm    = wave;                 // tile row owned by this wave

  // ---- CB[i][j] = sum_s C[i][s]*B[j][s]  (keep 4 tiles in registers)
  v8f accCB[4];
  #pragma unroll
  for (int tn = 0; tn < 4; ++tn) {
    v8f acc = {};
    const float* pA = &sC[(tm * 16 + lrow) * LDL];
    const float* pB = &sB[(tn * 16 + lrow) * LDL];
    #pragma unroll
    for (int kb = 0; kb < 16; ++kb) {
      const int k0 = kb * 4 + khalf;
      v2f a;  a.x = pA[k0];  a.y = pA[k0 + 1];
      v2f bb; bb.x = pB[k0]; bb.y = pB[k0 + 1];
      acc = wmma4(a, bb, acc);
    }
    accCB[tn] = acc;
  }

  // ---- h_final[s][p] = sum_c dte[c]*B[c][s]*X[c][p]  -> workspace
  float* hfc = hf + ((long)bh * NCH + nchunk) * (SD * PD);
  #pragma unroll
  for (int tn = 0; tn < 4; ++tn) {
    v8f acc = {};
    #pragma unroll
    for (int kb = 0; kb < 16; ++kb) {
      const int k0 = kb * 4 + khalf;
      v2f a, bb;
      a.x  = sDte[k0]     * sB[k0 * LDL       + tm * 16 + lrow];
      a.y  = sDte[k0 + 1] * sB[(k0 + 1) * LDL + tm * 16 + lrow];
      bb.x = sX[k0 * LDL       + tn * 16 + lrow];
      bb.y = sX[(k0 + 1) * LDL + tn * 16 + lrow];
      acc = wmma4(a, bb, acc);
    }
    #pragma unroll
    for (int r = 0; r < 8; ++r) {
      const int s = tm * 16 + mhalf + r;
      const int p = tn * 16 + lrow;
      hfc[s * PD + p] = acc[r];
    }
  }
  __syncthreads();    // all waves done reading sC

  // ---- overwrite sC with M = Lmat o CB
  #pragma unroll
  for (int tn = 0; tn < 4; ++tn) {
    #pragma unroll
    for (int r = 0; r < 8; ++r) {
      const int gi = tm * 16 + mhalf + r;
      const int gj = tn * 16 + lrow;
      float lm = 0.f;
      if (gi >= gj) {
        float d = sAcs[gi] - sAcs[gj];
        d = fminf(fmaxf(d, -100.f), 100.f);
        lm = expf(d);
      }
      sC[gi * LDL + gj] = accCB[tn][r] * lm;
    }
  }
  __syncthreads();

  // ---- Y_intra = M @ X  -> d_out
  float* gY = Y + row0 * PD;
  #pragma unroll
  for (int tn = 0; tn < 4; ++tn) {
    v8f acc = {};
    const float* pA = &sC[(tm * 16 + lrow) * LDL];
    #pragma unroll
    for (int kb = 0; kb < 16; ++kb) {
      const int k0 = kb * 4 + khalf;
      v2f a;  a.x = pA[k0]; a.y = pA[k0 + 1];
      v2f bb;
      bb.x = sX[k0 * LDL       + tn * 16 + lrow];
      bb.y = sX[(k0 + 1) * LDL + tn * 16 + lrow];
      acc = wmma4(a, bb, acc);
    }
    #pragma unroll
    for (int r = 0; r < 8; ++r) {
      const int gi = tm * 16 + mhalf + r;
      const int gj = tn * 16 + lrow;
      gY[(long)gi * (HN * PD) + gj] = acc[r];
    }
  }
}

// ---------------------------------------------------------------------------
// Kernel 2: sequential inter-chunk scan, in place: h_final -> h_inter
// h_inter[c] = state BEFORE chunk c;  carry = d_c*carry + h_final[c]
// Parallel over the 131072 independent (b,h,s,p) recurrences:
// 512 blocks x 256 threads, one state element per thread, coalesced.
// ---------------------------------------------------------------------------
__global__ __launch_bounds__(256)
void ssd_scan_kernel(float* __restrict__ hf, const float* __restrict__ acsbuf)
{
  __shared__ float sDc[NCH];
  const int blk = blockIdx.x;               // 0 .. 511
  const int bh  = blk >> 4;                 // 0..31
  const int seg = blk & 15;                 // 16 segments of 256 elements
  const int tid = threadIdx.x;
  if (tid < NCH)
    sDc[tid] = expf(acsbuf[((long)bh * NCH + tid) * CS + (CS - 1)]);
  __syncthreads();

  const int e = seg * 256 + tid;            // element of the 64x64 state
  float* base = hf + (long)bh * NCH * (SD * PD) + e;
  float carry = 0.f;
  #pragma unroll 4
  for (int c = 0; c < NCH; ++c) {
    float* pc = base + (long)c * (SD * PD);
    const float hv = *pc;
    *pc = carry;                            // h_inter[c] = state before chunk c
    carry = sDc[c] * carry + hv;
  }
}

// ---------------------------------------------------------------------------
// Kernel 3: Y += dfs[i] * (C[i,:] @ h_inter)   per (b,h,chunk)
// ---------------------------------------------------------------------------
__global__ __launch_bounds__(128)
void ssd_yinter_kernel(const float* __restrict__ Cm, const float* __restrict__ hf,
                       const float* __restrict__ acsbuf, float* __restrict__ Y)
{
  __shared__ float sC[CS * LDL];
  __shared__ float sH[SD * LDL];
  __shared__ float sDfs[CS];

  const int blk    = blockIdx.x;
  const int nchunk = blk & (NCH - 1);
  const int bh     = blk >> 6;
  const int h      = bh & (HN - 1);
  const int b      = bh >> 4;

  const int tid = threadIdx.x;
  const long row0 = (long)(b * LSEQ + nchunk * CS) * HN + h;
  const float* gC = Cm + row0 * SD;
  const float* gH = hf + ((long)bh * NCH + nchunk) * (SD * PD);

  for (int f = tid; f < CS * 16; f += 128) {
    const int row = f >> 4;
    const int c4  = (f & 15) << 2;
    const float4 cv = *(const float4*)(gC + (long)row * (HN * SD) + c4);
    const float4 hv = *(const float4*)(gH + (long)row * PD + c4);
    *(float4*)(&sC[row * LDL + c4]) = cv;
    *(float4*)(&sH[row * LDL + c4]) = hv;
  }
  if (tid < CS)
    sDfs[tid] = expf(acsbuf[((long)bh * NCH + nchunk) * CS + tid]);
  __syncthreads();

  const int lane  = tid & 31;
  const int wave  = tid >> 5;
  const int lrow  = lane & 15;
  const int khalf = (lane >> 4) << 1;
  const int mhalf = (lane >> 4) << 3;
  const int tm    = wave;

  float* gY = Y + row0 * PD;
  #pragma unroll
  for (int tn = 0; tn < 4; ++tn) {
    v8f acc = {};
    const float* pA = &sC[(tm * 16 + lrow) * LDL];
    #pragma unroll
    for (int kb = 0; kb < 16; ++kb) {
      const int k0 = kb * 4 + khalf;
      v2f a;  a.x = pA[k0]; a.y = pA[k0 + 1];
      v2f bb;
      bb.x = sH[k0 * LDL       + tn * 16 + lrow];
      bb.y = sH[(k0 + 1) * LDL + tn * 16 + lrow];
      acc = wmma4(a, bb, acc);
    }
    #pragma unroll
    for (int r = 0; r < 8; ++r) {
      const int gi = tm * 16 + mhalf + r;
      const int gj = tn * 16 + lrow;
      gY[(long)gi * (HN * PD) + gj] += sDfs[gi] * acc[r];
    }
  }
}

// ---------------------------------------------------------------------------
extern "C" void kernel_launch(void* const* d_in, const int* in_sizes, int n_in,
                              void* d_out, int out_size, void* d_ws, size_t ws_size,
                              hipStream_t stream) {
  const float* X  = (const float*)d_in[0];   // [2,4096,16,64]
  const float* A  = (const float*)d_in[1];   // [2,4096,16]
  const float* Bm = (const float*)d_in[2];   // [2,4096,16,64]
  const float* Cm = (const float*)d_in[3];   // [2,4096,16,64]
  float* Y = (float*)d_out;                  // [2,4096,16,64]

  float* hf     = (float*)d_ws;                               // 8,388,608 floats
  float* acsbuf = hf + (size_t)BATCH * HN * NCH * SD * PD;    // 131,072 floats

  const int nblocks = BATCH * HN * NCH;      // 2048
  ssd_chunk_kernel<<<nblocks, 128, 0, stream>>>(X, A, Bm, Cm, Y, hf, acsbuf);
  ssd_scan_kernel<<<BATCH * HN * 16, 256, 0, stream>>>(hf, acsbuf);
  ssd_yinter_kernel<<<nblocks, 128, 0, stream>>>(Cm, hf, acsbuf, Y);
}